// Model_82179904242122
// MI455X (gfx1250) — compile-verified
//
#include <hip/hip_runtime.h>
#include <stdint.h>

// Problem constants (from the reference)
#define B_      16
#define T_      1500
#define V_      10026      // VOCAB + 1 (logits depth)
#define BLANK_  10025
#define MASK_   10027      // VOCAB + 2
#define D_      512

// Async-staging geometry for the argmax kernel
#define CHUNK_FLOATS 1024                 // 256 lanes x 16 B
#define NCHUNK       10                   // ceil(10026 / 1024)
#define FULL_TAIL    10024                // floats [0,10024) covered by b128 copies
#define TAIL_LANES   202                  // lanes with a full valid float4 in chunk 9

// Branchless (max, first-index) comparator over a float4.
__device__ __forceinline__ void amax4(float4 q, int gbase, float& bv, int& bi) {
  float vals[4] = {q.x, q.y, q.z, q.w};
#pragma unroll
  for (int k = 0; k < 4; ++k) {
    int gidx = gbase + k;
    bool better = (vals[k] > bv) || (vals[k] == bv && gidx < bi);
    bv = better ? vals[k] : bv;
    bi = better ? gidx : bi;
  }
}

// ---------------------------------------------------------------------------
// Kernel 1: per-(b,t) argmax over V_ logits.
// One 256-thread block per row. Logits stream HBM->LDS via per-lane async
// b128 copies, 4-deep pipelined on ASYNCcnt. Each thread consumes only the
// LDS slot it copied itself, so the streaming loop needs NO barriers: the
// wave's own s_wait_asynccnt fully orders write->read.
// ---------------------------------------------------------------------------
__global__ __launch_bounds__(256) void ctc_argmax_kernel(
    const float* __restrict__ logits, int* __restrict__ preds) {
  const int row = blockIdx.x;             // 0 .. B_*T_-1
  const int tid = threadIdx.x;
  const float* rowp = logits + (size_t)row * V_;

  __shared__ float buf[4][CHUNK_FLOATS];  // 4 x 4KB staging ring (thread-private slots)
  __shared__ float red_v[8];
  __shared__ int   red_i[8];

  // LDS byte offsets (low 32 bits of the generic pointer are the wave-relative
  // LDS offset under the gfx1250 flat-aperture scheme).
  const unsigned ldsbase0 = (unsigned)(uintptr_t)&buf[0][0];
  const unsigned ldsbase1 = (unsigned)(uintptr_t)&buf[1][0];
  const unsigned ldsbase2 = (unsigned)(uintptr_t)&buf[2][0];
  const unsigned ldsbase3 = (unsigned)(uintptr_t)&buf[3][0];
  const unsigned ldsb[4] = {ldsbase0, ldsbase1, ldsbase2, ldsbase3};

#define ISSUE(c) do {                                                         \
    int base_f = (c) * CHUNK_FLOATS + tid * 4;                                \
    /* Clamp OOB lanes so EXEC stays all-ones and every wave's ASYNCcnt    */ \
    /* advances identically; garbage slots are never consumed.             */ \
    int safe = (base_f + 4 <= V_) ? base_f : 0;                               \
    unsigned long long ga = (unsigned long long)(uintptr_t)(rowp + safe);     \
    unsigned ldsoff = ldsb[(c) & 3] + (unsigned)tid * 16u;                    \
    asm volatile("global_load_async_to_lds_b128 %0, %1, off"                  \
                 :: "v"(ldsoff), "v"(ga) : "memory");                         \
  } while (0)

#define CONSUME(c) do {                                                       \
    float4 q = ((const float4*)&buf[(c) & 3][0])[tid];                        \
    amax4(q, (c) * CHUNK_FLOATS + tid * 4, bv, bi);                           \
  } while (0)

  float bv = -3.402823466e38f;
  int   bi = 0x7fffffff;

  // Prime 4 chunks (ASYNCcnt = 4 per wave).
  ISSUE(0); ISSUE(1); ISSUE(2); ISSUE(3);

  // Steady state: wait for chunk c (<=3 outstanding), consume it, then reuse
  // its buffer for chunk c+4. Reissue strictly after consume -> no hazard.
  asm volatile("s_wait_asynccnt 3" ::: "memory"); CONSUME(0); ISSUE(4);
  asm volatile("s_wait_asynccnt 3" ::: "memory"); CONSUME(1); ISSUE(5);
  asm volatile("s_wait_asynccnt 3" ::: "memory"); CONSUME(2); ISSUE(6);
  asm volatile("s_wait_asynccnt 3" ::: "memory"); CONSUME(3); ISSUE(7);
  asm volatile("s_wait_asynccnt 3" ::: "memory"); CONSUME(4); ISSUE(8);
  asm volatile("s_wait_asynccnt 3" ::: "memory"); CONSUME(5); ISSUE(9);
  // Drain.
  asm volatile("s_wait_asynccnt 3" ::: "memory"); CONSUME(6);
  asm volatile("s_wait_asynccnt 2" ::: "memory"); CONSUME(7);
  asm volatile("s_wait_asynccnt 1" ::: "memory"); CONSUME(8);
  asm volatile("s_wait_asynccnt 0" ::: "memory");
  // Chunk 9 is partial: lanes [0,202) hold 4 fully-valid floats (<= idx 10023).
  if (tid < TAIL_LANES) {
    float4 q = ((const float4*)&buf[9 & 3][0])[tid];
    amax4(q, 9 * CHUNK_FLOATS + tid * 4, bv, bi);
  }
  // Last two elements (10024, 10025): partial 16B, loaded directly.
  if (tid < 2) {
    int gidx = FULL_TAIL + tid;
    float v = rowp[gidx];
    bool better = (v > bv) || (v == bv && gidx < bi);
    bv = better ? v : bv;
    bi = better ? gidx : bi;
  }
#undef ISSUE
#undef CONSUME

  // wave32 reduction (first-occurrence tie-break to match jnp.argmax)
#pragma unroll
  for (int off = 16; off > 0; off >>= 1) {
    float ov = __shfl_down(bv, off, 32);
    int   oi = __shfl_down(bi, off, 32);
    bool better = (ov > bv) || (ov == bv && oi < bi);
    bv = better ? ov : bv;
    bi = better ? oi : bi;
  }
  if ((tid & 31) == 0) { red_v[tid >> 5] = bv; red_i[tid >> 5] = bi; }
  __syncthreads();                        // only barrier in the kernel
  if (tid == 0) {
#pragma unroll
    for (int w = 1; w < 8; ++w) {
      float ov = red_v[w]; int oi = red_i[w];
      bool better = (ov > bv) || (ov == bv && oi < bi);
      bv = better ? ov : bv;
      bi = better ? oi : bi;
    }
    preds[row] = bi;
  }
}

// ---------------------------------------------------------------------------
// Kernel 2: per-batch run-collapse + blank-drop + compaction (block scan).
// One 256-thread block per batch row; 6 elements per thread.
// ---------------------------------------------------------------------------
__global__ __launch_bounds__(256) void ctc_collapse_kernel(
    const int* __restrict__ preds, const int* __restrict__ audio_len,
    int* __restrict__ tokens, int* __restrict__ lens,
    float* __restrict__ lens_out_f) {
  const int b = blockIdx.x;
  const int tid = threadIdx.x;
  __shared__ int sp[T_];
  __shared__ int ssum[256];

  const int len = audio_len[b];
  for (int t = tid; t < T_; t += 256) sp[t] = preds[b * T_ + t];
  for (int t = tid; t < T_; t += 256) tokens[b * T_ + t] = 0;
  __syncthreads();

  int cnt = 0;
  unsigned keepmask = 0;
#pragma unroll
  for (int k = 0; k < 6; ++k) {
    int t = tid * 6 + k;
    if (t < T_) {
      int p = sp[t];
      bool keep = (t < len) && (p != BLANK_) && (t == 0 || p != sp[t - 1]);
      if (keep) { keepmask |= (1u << k); ++cnt; }
    }
  }

  ssum[tid] = cnt;
  __syncthreads();
  for (int off = 1; off < 256; off <<= 1) {   // inclusive Hillis-Steele scan
    int add = (tid >= off) ? ssum[tid - off] : 0;
    __syncthreads();
    ssum[tid] += add;
    __syncthreads();
  }
  const int total = ssum[255];
  int pos = ssum[tid] - cnt;                  // exclusive prefix

#pragma unroll
  for (int k = 0; k < 6; ++k) {
    int t = tid * 6 + k;
    if (t < T_ && (keepmask & (1u << k))) {
      tokens[b * T_ + pos] = sp[t];
      ++pos;
    }
  }
  __syncthreads();
  if (tid == 0) {
    int L = total;
    if (L == 0) { tokens[b * T_] = MASK_; L = 1; }
    lens[b] = L;
    lens_out_f[b] = (float)L;                 // second reference output, as f32
  }
}

// ---------------------------------------------------------------------------
// Kernel 3: masked embedding gather, float4-vectorized. Table (~20.5 MB) is
// L2-resident on the 192 MB L2, output writes are fully coalesced.
// ---------------------------------------------------------------------------
__global__ __launch_bounds__(256) void ctc_embed_kernel(
    const float* __restrict__ emb_table, const int* __restrict__ tokens,
    const int* __restrict__ lens, float* __restrict__ out) {
  const int D4 = D_ / 4;                      // 128 float4 per row
  long long i = (long long)blockIdx.x * blockDim.x + threadIdx.x;
  const long long total = (long long)B_ * T_ * D4;
  if (i >= total) return;
  int d4  = (int)(i & (D4 - 1));
  int row = (int)(i >> 7);
  int b = row / T_;
  int t = row - b * T_;
  float4 v = make_float4(0.f, 0.f, 0.f, 0.f);
  if (t < lens[b]) {
    int tok = tokens[row];
    v = ((const float4*)emb_table)[(size_t)tok * D4 + d4];
  }
  ((float4*)out)[i] = v;
}

// ---------------------------------------------------------------------------
extern "C" void kernel_launch(void* const* d_in, const int* in_sizes, int n_in,
                              void* d_out, int out_size, void* d_ws, size_t ws_size,
                              hipStream_t stream) {
  (void)in_sizes; (void)n_in; (void)out_size; (void)ws_size;
  const float* logits    = (const float*)d_in[0];   // [16,1500,10026] f32
  const float* emb_table = (const float*)d_in[1];   // [10028,512]     f32
  const int*   audio_len = (const int*)d_in[2];     // [16]            i32
  float* out = (float*)d_out;                       // [16*1500*512] emb + [16] lens

  int* preds  = (int*)d_ws;                         // B*T
  int* tokens = preds + B_ * T_;                    // B*T
  int* lens   = tokens + B_ * T_;                   // B

  ctc_argmax_kernel<<<B_ * T_, 256, 0, stream>>>(logits, preds);
  ctc_collapse_kernel<<<B_, 256, 0, stream>>>(preds, audio_len, tokens, lens,
                                              out + (size_t)B_ * T_ * D_);
  const int total4 = B_ * T_ * (D_ / 4);
  ctc_embed_kernel<<<(total4 + 255) / 256, 256, 0, stream>>>(emb_table, tokens,
                                                             lens, out);
}